// PinnLoss_58299886076550
// MI455X (gfx1250) — compile-verified
//
#include <hip/hip_runtime.h>
#include <hip/hip_bf16.h>

// ---------------- problem constants (match reference) ----------------
#define BSZ        32768
#define TSEQ       240
#define CCH        8
#define NODE       100
#define T_MAX_H    120.0f
#define T_MIN_H    12.0f
#define GAMMA0     5e-08f
#define NREF       5.0f
#define R_START_KM 14964000.0f    // 21.5 * 696000
#define R_STOP_KM  149640000.0f   // 215  * 696000
#define DT_HRS     1.2f           // 120/100
#define DT_SEC     4320.0f        // 1.2*3600
#define L_ODE_W    1.0f
#define L_MONO_W   1.0f
#define L_BOUND_W  0.1f
#define L_QORD_W   0.5f

typedef __attribute__((ext_vector_type(2))) float v2f;
typedef __attribute__((ext_vector_type(4))) float v4f;
typedef __attribute__((ext_vector_type(8))) float v8f;

// ws layout (floats): [0,4096) mono per-block partials
//                     [4096,4608) ode-kernel partials: 4 arrays of 128
//                     [4608, 4608+B*T) gamma staging (optional)
#define WS_PART_MONO 0
#define WS_PART_K2   4096
#define WS_GAMMA     4608

// ---------------- wave reduction via the matrix pipe ----------------
// Sum of all 32 lanes' v. A[m][0]=v_m, A[m][2]=v_{m+16}, B=ones(4x16):
// D[m][n] = v_m + v_{m+16}.  Lane q<16 holds rows 0..7 of D, lane q>=16 rows
// 8..15; summing the 8 D VGPRs gives the half-sum, one shfl finishes it.
// Result valid on lane 0 (lanes 0..15 in the WMMA path).
__device__ __forceinline__ float wave_reduce_sum(float v) {
#if __has_builtin(__builtin_amdgcn_wmma_f32_16x16x4_f32)
  v2f a; a.x = v;   a.y = 0.0f;
  v2f b; b.x = 1.0f; b.y = 1.0f;          // all-ones B under any layout
  v8f c = {};
  v8f d = __builtin_amdgcn_wmma_f32_16x16x4_f32(
      false, a, false, b, (short)0, c, false, false);
  float s = d[0] + d[1] + d[2] + d[3] + d[4] + d[5] + d[6] + d[7];
  s += __shfl_down(s, 16);                // add rows 8..15 half
  return s;
#else
  for (int off = 16; off > 0; off >>= 1) v += __shfl_down(v, off);
  return v;
#endif
}

// Deterministic block reduction (blockDim.x multiple of 32, <=256 here).
__device__ __forceinline__ float block_reduce_sum(float v, float* sm) {
  int lane = threadIdx.x & 31;
  int w    = threadIdx.x >> 5;
  float ws = wave_reduce_sum(v);
  if (lane == 0) sm[w] = ws;
  __syncthreads();
  float r = 0.0f;
  if (w == 0) {
    float t = (lane < (int)(blockDim.x >> 5)) ? sm[lane] : 0.0f;
    r = wave_reduce_sum(t);               // wave 0 fully active: EXEC all 1s
  }
  __syncthreads();
  return r;                               // valid on thread 0
}

// ---------------- kernel 1: stream x_seq/m_seq once ----------------
// 8 waves/block, one sample per wave, lanes cover 32 consecutive t:
// coalesced 16B/lane NT loads. Computes mono-loss partials and stages
// gamma(b,t) into L2-resident workspace for the ODE kernel.
__global__ void __launch_bounds__(256)
stream_kernel(const float* __restrict__ x, const float* __restrict__ m,
              const float* __restrict__ v_sw,
              float* __restrict__ part_mono, float* __restrict__ gamma) {
  __shared__ float sm[8];
  const int lane = threadIdx.x & 31;
  const int w    = threadIdx.x >> 5;
  const int b    = blockIdx.x * 8 + w;

  const v4f* __restrict__ x4 = reinterpret_cast<const v4f*>(x);
  const v4f* __restrict__ m4 = reinterpret_cast<const v4f*>(m);

  const float vs = v_sw[b];
  float acc = 0.0f;

  for (int c = 0; c < 8; ++c) {
    const int t = c * 32 + lane;                       // 0..255
    v4f xv = {}; v4f mv = {};
    if (t < TSEQ) {
      xv = __builtin_nontemporal_load(&x4[(size_t)b * TSEQ + t]);
      mv = __builtin_nontemporal_load(&m4[(size_t)b * TSEQ + t]);
      if (gamma) {                                     // channel 2 = density
        float ds = fmaxf(xv.z * mv.z + NREF * (1.0f - mv.z), 0.01f);
        float g  = GAMMA0 * ds * (1.0f / NREF);
        g = fminf(fmaxf(g, 1e-9f), 1e-6f);
        gamma[(size_t)b * TSEQ + t] = g;
      }
    }
    // neighbor (t+1) speed/mask from the wave, lane31 from cache
    float sn = __shfl_down(xv.y, 1);
    float mn = __shfl_down(mv.y, 1);
    if (t < TSEQ - 1) {
      if (lane == 31) {
        size_t g1 = ((size_t)b * TSEQ + t + 1) * CCH + 1;   // channel 1
        sn = x[g1];
        mn = m[g1];
      }
      float mp    = mv.y * mn;
      float above = (xv.y > vs) ? 1.0f : 0.0f;
      float inc   = fmaxf(sn - xv.y, 0.0f);
      float viol  = above * inc * mp;
      acc += viol * viol;
    }
  }

  float tot = block_reduce_sum(acc, sm);
  if (threadIdx.x == 0) part_mono[blockIdx.x] = tot;
}

// ---------------- kernel 2: per-sample ODE scan + small losses -------
__global__ void __launch_bounds__(256)
ode_kernel(const float* __restrict__ preds, const float* __restrict__ targets,
           const float* __restrict__ v0_kms, const float* __restrict__ v_sw,
           const float* __restrict__ x, const float* __restrict__ m,
           const float* __restrict__ gamma, float* __restrict__ part2) {
  __shared__ float sm[8];
  const int b = blockIdx.x * blockDim.x + threadIdx.x;   // exactly BSZ threads

  const float p0 = preds[3 * b + 0];
  const float p1 = preds[3 * b + 1];
  const float p2 = preds[3 * b + 2];
  const float tg = targets[b];
  const float vs = v_sw[b];

  // pinball: mean over (B,3) of max(q*err,(q-1)*err)
  float pin = 0.0f;
  {
    const float q[3]  = {0.1f, 0.5f, 0.9f};
    const float pv[3] = {p0, p1, p2};
    #pragma unroll
    for (int j = 0; j < 3; ++j) {
      float err = tg - pv[j];
      pin += fmaxf(q[j] * err, (q[j] - 1.0f) * err);
    }
  }
  // bounds: relu(12-p)^2 + relu(p-120)^2 over 3 quantiles
  float bnd = 0.0f;
  {
    const float pv[3] = {p0, p1, p2};
    #pragma unroll
    for (int j = 0; j < 3; ++j) {
      float lo = fmaxf(T_MIN_H - pv[j], 0.0f);
      float hi = fmaxf(pv[j] - T_MAX_H, 0.0f);
      bnd += lo * lo + hi * hi;
    }
  }
  // quantile ordering
  float v1 = fmaxf(p0 - p1, 0.0f);
  float v2 = fmaxf(p1 - p2, 0.0f);
  float qo = v1 * v1 + v2 * v2;

  // drag ODE: 100 sequential steps (gamma from L2-resident staging)
  float v = v0_kms[b];
  float r = R_START_KM;
  float arrival = T_MAX_H;
  const float step = 239.0f / 99.0f;                    // linspace(0,239,100)
  #pragma unroll 4
  for (int k = 0; k < NODE; ++k) {
    int tk = (int)((float)k * step);
    tk = (tk < TSEQ - 1) ? tk : (TSEQ - 1);
    float g;
    if (gamma) {
      g = gamma[(size_t)b * TSEQ + tk];
    } else {
      size_t gi = ((size_t)b * TSEQ + tk) * CCH + 2;    // channel 2
      float d  = x[gi];
      float mm = m[gi];
      float ds = fmaxf(d * mm + NREF * (1.0f - mm), 0.01f);
      g = fminf(fmaxf(GAMMA0 * ds * (1.0f / NREF), 1e-9f), 1e-6f);
    }
    float dv = vs - v;
    float vn = v + g * fabsf(dv) * dv * DT_SEC;
    vn = (v > vs) ? fmaxf(vn, vs) : fminf(vn, vs);
    float rn = r + 0.5f * (v + vn) * DT_SEC;
    bool crossed = (r < R_STOP_KM) && (rn >= R_STOP_KM);
    float frac = fminf(fmaxf((R_STOP_KM - r) / (rn - r + 1e-12f), 0.0f), 1.0f);
    float tc = ((float)k + frac) * DT_HRS;
    if (crossed && (arrival >= T_MAX_H)) arrival = tc;
    v = vn; r = rn;
  }
  float ode = (p1 - arrival) * (p1 - arrival);

  // deterministic per-block partials (4 arrays of gridDim.x=128)
  float s;
  s = block_reduce_sum(pin, sm); if (threadIdx.x == 0) part2[0 * 128 + blockIdx.x] = s;
  s = block_reduce_sum(bnd, sm); if (threadIdx.x == 0) part2[1 * 128 + blockIdx.x] = s;
  s = block_reduce_sum(qo,  sm); if (threadIdx.x == 0) part2[2 * 128 + blockIdx.x] = s;
  s = block_reduce_sum(ode, sm); if (threadIdx.x == 0) part2[3 * 128 + blockIdx.x] = s;
}

// ---------------- kernel 3: deterministic finalize -------------------
__global__ void __launch_bounds__(256)
finalize_kernel(const float* __restrict__ ws, float* __restrict__ out) {
  __shared__ float sm[8];
  const int tid = threadIdx.x;

  float a = 0.0f;
  for (int i = tid; i < 4096; i += 256) a += ws[WS_PART_MONO + i];
  float mono_s = block_reduce_sum(a, sm);

  float v;
  v = (tid < 128) ? ws[WS_PART_K2 + 0 * 128 + tid] : 0.0f;
  float pin_s = block_reduce_sum(v, sm);
  v = (tid < 128) ? ws[WS_PART_K2 + 1 * 128 + tid] : 0.0f;
  float bnd_s = block_reduce_sum(v, sm);
  v = (tid < 128) ? ws[WS_PART_K2 + 2 * 128 + tid] : 0.0f;
  float qo_s = block_reduce_sum(v, sm);
  v = (tid < 128) ? ws[WS_PART_K2 + 3 * 128 + tid] : 0.0f;
  float ode_s = block_reduce_sum(v, sm);

  if (tid == 0) {
    const float L_mono  = mono_s / ((float)BSZ * (float)(TSEQ - 1));
    const float L_pin   = pin_s  / ((float)BSZ * 3.0f);
    const float L_bound = bnd_s  / ((float)BSZ * 3.0f);
    const float L_qord  = qo_s   / (float)BSZ;
    const float L_ode   = ode_s  / (float)BSZ;
    const float total = L_pin + L_BOUND_W * L_bound + L_QORD_W * L_qord
                      + L_ODE_W * L_ode + L_MONO_W * L_mono;
    out[0] = total;
    out[1] = L_pin;
    out[2] = L_ode;
    out[3] = L_mono;
    out[4] = L_bound;
    out[5] = L_qord;
  }
}

extern "C" void kernel_launch(void* const* d_in, const int* in_sizes, int n_in,
                              void* d_out, int out_size, void* d_ws, size_t ws_size,
                              hipStream_t stream) {
  (void)in_sizes; (void)n_in; (void)out_size;
  const float* preds   = (const float*)d_in[0];
  const float* targets = (const float*)d_in[1];
  const float* v0_kms  = (const float*)d_in[2];
  const float* v_sw    = (const float*)d_in[3];
  const float* x_seq   = (const float*)d_in[4];
  const float* m_seq   = (const float*)d_in[5];

  float* ws = (float*)d_ws;
  float* part_mono = ws + WS_PART_MONO;
  float* part2     = ws + WS_PART_K2;
  const size_t need = ((size_t)WS_GAMMA + (size_t)BSZ * TSEQ) * sizeof(float);
  float* gamma = (ws_size >= need) ? (ws + WS_GAMMA) : nullptr;

  stream_kernel<<<BSZ / 8, 256, 0, stream>>>(x_seq, m_seq, v_sw, part_mono, gamma);
  ode_kernel<<<BSZ / 256, 256, 0, stream>>>(preds, targets, v0_kms, v_sw,
                                            x_seq, m_seq, gamma, part2);
  finalize_kernel<<<1, 256, 0, stream>>>(ws, (float*)d_out);
}